// AttentionGraphLevel_83296595739229
// MI455X (gfx1250) — compile-verified
//
#include <hip/hip_runtime.h>
#include <math.h>

// ---------------------------------------------------------------------------
// AttentionGraphLevel on MI455X (gfx1250, wave32)
//
// Math collapses: alpha & f_out depend only on group id -> per-group scalars
//   A_g = sigmoid(x_g . w_w + w_b) * f_out_g ,  B_g = 1 - alpha_g
//   score_i = A_g * rowsum(x_i) + B_g * (x_i . attn_w)
//   out[g]  = sum_i softmax_g(score)_i * x_i   (global-max shift cancels)
//
// Memory-bound: node_embeddings = 256 MB. Online softmax per wave reads it
// exactly once -> ~11 us roofline at 23.3 TB/s.
// WMMA (V_WMMA_F32_16X16X4_F32) handles the per-group MLP GEMM
// [Bpad x 128] x [128 x 128].
// ---------------------------------------------------------------------------

#define C 128
#define NTILES (C / 16)   // 8 column tiles in the group GEMM
#define P2_WAVES 8
#define P2_THREADS (P2_WAVES * 32)

typedef __attribute__((ext_vector_type(2))) float v2f;
typedef __attribute__((ext_vector_type(8))) float v8f;

// ---- kernel 0: workspace init (harness does not re-zero ws between runs) ---
__global__ void ws_init_kernel(int* __restrict__ gstart, int* __restrict__ gend,
                               float* __restrict__ fout, int B, int N) {
  int i = blockIdx.x * blockDim.x + threadIdx.x;
  if (i < B) { gstart[i] = N; gend[i] = 0; fout[i] = 0.0f; }
}

// ---- kernel 1: group boundaries (groups sorted -> neighbor compare, no atomics)
__global__ void bounds_kernel(const int* __restrict__ groups,
                              int* __restrict__ gstart, int* __restrict__ gend,
                              int N) {
  int i = blockIdx.x * blockDim.x + threadIdx.x;
  if (i >= N) return;
  int g  = groups[i];
  int gn = (i + 1 < N) ? groups[i + 1] : -1;
  if (i == 0) gstart[g] = 0;
  if (gn != g) {
    gend[g] = i + 1;
    if (gn >= 0) gstart[gn] = i + 1;
  }
}

// ---- kernel 2: per-group MLP via f32 WMMA --------------------------------
// H = relu(L @ f1_w + f1_b); fout[g] = H[g,:] . f2_w
// One wave per 16x16 D tile; fully unrolled K-loop of v_wmma_f32_16x16x4_f32.
// A-frag layout (ISA 32-bit 16x4): lanes0-15: M=lane, VGPR0=K0,VGPR1=K1;
//                                  lanes16-31: M=lane-16, K2/K3.
// B-frag (4x16) symmetric: lanes0-15 col=lane hold K0/K1; lanes16-31 K2/K3.
// D layout: VGPR v -> row = v (+8 for upper half-lanes), col = lane&15.
//
// Padding rows (m >= B) are CLAMPED to row B-1 instead of zero-predicated:
// they only feed D rows >= B, which are discarded below. This keeps every
// load unconditional (no exec-mask save/restore around scalar loads) and
// lets the A fragment come in as a single global_load_b64.
__global__ void group_mlp_wmma(const float* __restrict__ lat,
                               const float* __restrict__ f1w,
                               const float* __restrict__ f1b,
                               const float* __restrict__ f2w,
                               float* __restrict__ fout, int B) {
  int mt   = (int)blockIdx.x >> 3;        // M tile
  int nt   = (int)blockIdx.x & 7;         // N tile
  int lane = (int)threadIdx.x;            // 0..31 (wave32)
  int half = lane >> 4;
  int l15  = lane & 15;
  int m    = mt * 16 + l15;               // A row handled by this lane
  int mcl  = (m < B) ? m : (B - 1);       // in-bounds clamp (see note above)
  int n    = nt * 16 + l15;               // B/D column handled by this lane
  int kb   = half * 2;                    // lane-half K base within each K4

  const float* aptr = lat + (size_t)mcl * C + kb;   // 8B-aligned (kb in {0,2}, k0%4==0)
  const float* bptr = f1w + (size_t)kb * C + n;

  v8f acc = {0.f, 0.f, 0.f, 0.f, 0.f, 0.f, 0.f, 0.f};
#pragma unroll
  for (int k0 = 0; k0 < C; k0 += 4) {
    v2f a = *(const v2f*)(aptr + k0);     // one b64 load: K=k0+kb, k0+kb+1
    v2f b;
    b.x = bptr[k0 * C];                   // F1[k0+kb,   n]
    b.y = bptr[k0 * C + C];               // F1[k0+kb+1, n]
    acc = __builtin_amdgcn_wmma_f32_16x16x4_f32(
        /*neg_a=*/false, a, /*neg_b=*/false, b,
        /*c_mod=*/(short)0, acc, /*reuse_a=*/false, /*reuse_b=*/false);
  }

  float bias = f1b[n];
  float w2   = f2w[n];
  // Each (VGPR v, half) holds one matrix row across 16 lanes -> half-wave
  // reductions give per-row partial sums over this 16-col tile.
#pragma unroll
  for (int v = 0; v < 8; ++v) {
    float h = fmaxf(acc[v] + bias, 0.0f) * w2;
    h += __shfl_xor(h, 1, 32);
    h += __shfl_xor(h, 2, 32);
    h += __shfl_xor(h, 4, 32);
    h += __shfl_xor(h, 8, 32);
    int row = mt * 16 + half * 8 + v;
    if (l15 == 0 && row < B) atomicAdd(&fout[row], h);
  }
}

// ---- kernel 3: per-group online-softmax weighted pooling ------------------
// One block per group; 8 waves; each wave streams whole node rows (float4 per
// lane, fully coalesced 512B/row) exactly once, keeping a flash-style running
// (max, denom, acc[4]) and merging waves through LDS at the end.
__global__ void __launch_bounds__(P2_THREADS)
group_pool_kernel(const float* __restrict__ X,      // node_embeddings [N,C]
                  const float* __restrict__ attn_w, // [C]
                  const float* __restrict__ w_w,    // [C]
                  const float* __restrict__ w_b,    // [1]
                  const float* __restrict__ f2_b,   // [1]
                  const int* __restrict__ gstart,
                  const int* __restrict__ gend,
                  const float* __restrict__ fout,   // [B]
                  float* __restrict__ out,          // [B,C]
                  int B) {
  int g    = blockIdx.x;
  int tid  = (int)threadIdx.x;
  int lane = tid & 31;
  int wave = tid >> 5;

  __shared__ float s_m[P2_WAVES];
  __shared__ float s_d[P2_WAVES];
  __shared__ float s_acc[P2_WAVES][C];
  __shared__ float s_par[2];   // A_g, B_g

  const float4* X4 = (const float4*)X;

  // Per-group scalars (wave 0): alpha = sigmoid(X[g] . w_w + w_b)
  if (wave == 0) {
    float4 cx = X4[g * (C / 4) + lane];
    float4 ww = ((const float4*)w_w)[lane];
    float p = cx.x * ww.x + cx.y * ww.y + cx.z * ww.z + cx.w * ww.w;
    p += __shfl_xor(p, 1, 32);
    p += __shfl_xor(p, 2, 32);
    p += __shfl_xor(p, 4, 32);
    p += __shfl_xor(p, 8, 32);
    p += __shfl_xor(p, 16, 32);
    if (lane == 0) {
      float alpha = 1.0f / (1.0f + __expf(-(p + w_b[0])));
      float fo    = fout[g] + f2_b[0];
      s_par[0] = alpha * fo;     // A_g
      s_par[1] = 1.0f - alpha;   // B_g
    }
  }
  __syncthreads();
  float Acoef = s_par[0];
  float Bcoef = s_par[1];
  float4 aw = ((const float4*)attn_w)[lane];

  int start = gstart[g];
  int end   = gend[g];

  float  m = -INFINITY;
  float  d = 0.0f;
  float4 acc = make_float4(0.f, 0.f, 0.f, 0.f);

  for (int j = start + wave; j < end; j += P2_WAVES) {
    float4 x = X4[j * (C / 4) + lane];
    float p1 = x.x + x.y + x.z + x.w;                               // rowsum
    float p2 = x.x * aw.x + x.y * aw.y + x.z * aw.z + x.w * aw.w;   // x.attn_w
    p1 += __shfl_xor(p1, 1, 32);  p2 += __shfl_xor(p2, 1, 32);
    p1 += __shfl_xor(p1, 2, 32);  p2 += __shfl_xor(p2, 2, 32);
    p1 += __shfl_xor(p1, 4, 32);  p2 += __shfl_xor(p2, 4, 32);
    p1 += __shfl_xor(p1, 8, 32);  p2 += __shfl_xor(p2, 8, 32);
    p1 += __shfl_xor(p1, 16, 32); p2 += __shfl_xor(p2, 16, 32);
    float s  = Acoef * p1 + Bcoef * p2;
    float mn = fmaxf(m, s);
    float sc = __expf(m - mn);    // 0 on first iter (exp(-inf))
    float pw = __expf(s - mn);
    d = d * sc + pw;
    acc.x = acc.x * sc + pw * x.x;
    acc.y = acc.y * sc + pw * x.y;
    acc.z = acc.z * sc + pw * x.z;
    acc.w = acc.w * sc + pw * x.w;
    m = mn;
  }

  if (lane == 0) { s_m[wave] = m; s_d[wave] = d; }
  s_acc[wave][lane * 4 + 0] = acc.x;
  s_acc[wave][lane * 4 + 1] = acc.y;
  s_acc[wave][lane * 4 + 2] = acc.z;
  s_acc[wave][lane * 4 + 3] = acc.w;
  __syncthreads();

  if (tid < C) {
    float M = -INFINITY;
    for (int w = 0; w < P2_WAVES; ++w) M = fmaxf(M, s_m[w]);
    if (!(M > -INFINITY)) {            // empty group -> zeros (matches ref)
      out[g * C + tid] = 0.0f;
      return;
    }
    float denom = 0.0f, val = 0.0f;
    for (int w = 0; w < P2_WAVES; ++w) {
      float e = __expf(s_m[w] - M);    // exp(-inf)=0 for idle waves
      denom += s_d[w] * e;
      val   += s_acc[w][tid] * e;
    }
    out[g * C + tid] = val / denom;
  }
}

// ---------------------------------------------------------------------------
extern "C" void kernel_launch(void* const* d_in, const int* in_sizes, int n_in,
                              void* d_out, int out_size, void* d_ws, size_t ws_size,
                              hipStream_t stream) {
  (void)n_in; (void)out_size; (void)ws_size;
  const float* node_emb = (const float*)d_in[0];   // [N,C]
  const float* lat      = (const float*)d_in[1];   // [B,C]
  const float* attn_w   = (const float*)d_in[2];   // [C]
  const float* w_w      = (const float*)d_in[3];   // [C]
  const float* w_b      = (const float*)d_in[4];   // [1]
  const float* f1_w     = (const float*)d_in[5];   // [C,C]
  const float* f1_b     = (const float*)d_in[6];   // [C]
  const float* f2_w     = (const float*)d_in[7];   // [C]
  const float* f2_b     = (const float*)d_in[8];   // [1]
  const int*   groups   = (const int*)d_in[9];     // [N]

  int N = in_sizes[9];
  int B = in_sizes[1] / C;

  // Workspace: [int start[B] | int end[B] | float fout[B]]
  int*   gstart = (int*)d_ws;
  int*   gend   = gstart + B;
  float* fout   = (float*)(gend + B);

  ws_init_kernel<<<(B + 255) / 256, 256, 0, stream>>>(gstart, gend, fout, B, N);
  bounds_kernel<<<(N + 255) / 256, 256, 0, stream>>>(groups, gstart, gend, N);

  int mtiles = (B + 15) / 16;
  group_mlp_wmma<<<mtiles * NTILES, 32, 0, stream>>>(lat, f1_w, f1_b, f2_w, fout, B);

  group_pool_kernel<<<B, P2_THREADS, 0, stream>>>(node_emb, attn_w, w_w, w_b, f2_b,
                                                  gstart, gend, fout,
                                                  (float*)d_out, B);
}